// GraphAttentionLayer_63101659513099
// MI455X (gfx1250) — compile-verified
//
#include <hip/hip_runtime.h>
#include <hip/hip_bf16.h>

typedef float v2f __attribute__((ext_vector_type(2)));
typedef float v8f __attribute__((ext_vector_type(8)));

#define IN_CH 128
#define HC 64          // HEADS*OUT_CH
#define OUT_CH 32
#define NEG_SLOPE 0.2f

// -------- monotone float <-> uint key for atomic max --------
__device__ __forceinline__ unsigned fkey(float f) {
    unsigned u = __float_as_uint(f);
    return (u & 0x80000000u) ? ~u : (u | 0x80000000u);
}
__device__ __forceinline__ float fdec(unsigned k) {
    unsigned u = (k & 0x80000000u) ? (k & 0x7fffffffu) : ~k;
    return __uint_as_float(u);
}

// ============================================================
// Kernel 1: xlr[N][128] = x[N][128] @ [Wl | Wr] + [bl | br]
// One wave computes a 16x16 tile via V_WMMA_F32_16X16X4_F32.
// 8 waves per block cover all 8 column tiles of one row tile.
// ============================================================
__global__ __launch_bounds__(256) void gemm_xlr(
    const float* __restrict__ x,
    const float* __restrict__ Wl, const float* __restrict__ Wr,
    const float* __restrict__ bl, const float* __restrict__ br,
    float* __restrict__ xlr)
{
    const int lane = threadIdx.x & 31;
    const int wave = threadIdx.x >> 5;     // 0..7 -> N tile
    const int m0   = blockIdx.x * 16;      // row tile base
    const int n0   = wave * 16;            // col tile base (0..112)
    const int half = lane >> 4;            // 0: K={0,1}, 1: K={2,3}
    const int l15  = lane & 15;

    v8f acc = {};
    const int arow = m0 + l15;             // A row for this lane
    const int bn   = n0 + l15;             // B column for this lane
    const float* Bp = (bn < HC) ? Wl : Wr;
    const int    bj = (bn < HC) ? bn : bn - HC;

    for (int k = 0; k < IN_CH; k += 4) {
        const int kk = k + half * 2;
        v2f a, b;
        a.x = x[arow * IN_CH + kk];
        a.y = x[arow * IN_CH + kk + 1];
        b.x = Bp[kk * HC + bj];
        b.y = Bp[(kk + 1) * HC + bj];
        acc = __builtin_amdgcn_wmma_f32_16x16x4_f32(
            false, a, false, b, (short)0, acc, false, false);
    }

    const float bv = (bn < HC) ? bl[bn] : br[bn - HC];
#pragma unroll
    for (int v = 0; v < 8; ++v) {
        const int row = m0 + half * 8 + v;
        xlr[row * IN_CH + bn] = acc[v] + bv;
    }
}

// ============================================================
// Kernel 2: per-edge attention logit + atomic segment max.
// One wave per edge. lane -> head = lane>>4, chans (lane&15),(lane&15)+16
// ============================================================
__global__ __launch_bounds__(256) void edge_logits(
    const int* __restrict__ ei, const float* __restrict__ xlr,
    const float* __restrict__ att, float* __restrict__ ebuf,
    unsigned* __restrict__ mkey, int nE, int nN)
{
    const long long eTot = (long long)nE + nN;
    const long long e = (long long)blockIdx.x * 8 + (threadIdx.x >> 5);
    if (e >= eTot) return;
    const int lane = threadIdx.x & 31;

    int src, dst;
    if (e < nE) { src = ei[e]; dst = ei[(long long)nE + e]; }
    else        { src = (int)(e - nE); dst = src; }

    const int h = lane >> 4;
    const int c = lane & 15;
    const float* xl = xlr + (long long)src * IN_CH + h * OUT_CH;
    const float* xr = xlr + (long long)dst * IN_CH + HC + h * OUT_CH;

    float s0 = xl[c]      + xr[c];
    float s1 = xl[c + 16] + xr[c + 16];
    s0 = (s0 > 0.f) ? s0 : NEG_SLOPE * s0;
    s1 = (s1 > 0.f) ? s1 : NEG_SLOPE * s1;
    float p = s0 * att[h * OUT_CH + c] + s1 * att[h * OUT_CH + c + 16];

    p += __shfl_xor(p, 8, 32);
    p += __shfl_xor(p, 4, 32);
    p += __shfl_xor(p, 2, 32);
    p += __shfl_xor(p, 1, 32);   // each 16-lane half now holds its head sum

    if (c == 0) {
        ebuf[e * 2 + h] = p;
        atomicMax(&mkey[(long long)dst * 2 + h], fkey(p));
    }
}

// ============================================================
// Kernel 3: a = exp(e - max[dst]); denom[dst] += a (thread per edge-head)
// ============================================================
__global__ __launch_bounds__(256) void edge_exp(
    const int* __restrict__ ei, float* __restrict__ ebuf,
    const unsigned* __restrict__ mkey, float* __restrict__ denom,
    int nE, int nN)
{
    const long long eTot = (long long)nE + nN;
    const long long t = (long long)blockIdx.x * blockDim.x + threadIdx.x;
    if (t >= eTot * 2) return;
    const long long e = t >> 1;
    const int h = (int)(t & 1);
    const int dst = (e < nE) ? ei[(long long)nE + e] : (int)(e - nE);
    const float m = fdec(mkey[(long long)dst * 2 + h]);
    const float a = __expf(ebuf[t] - m);
    ebuf[t] = a;
    atomicAdd(&denom[(long long)dst * 2 + h], a);
}

// ============================================================
// Kernel 4: out[dst][c] += (alpha_h / 2) * xl[src][h][c]  (wave per edge)
// Head-mean folded into the 0.5 scale; both heads land on same column.
// ============================================================
__global__ __launch_bounds__(256) void edge_scatter(
    const int* __restrict__ ei, const float* __restrict__ xlr,
    const float* __restrict__ ebuf, const float* __restrict__ denom,
    float* __restrict__ out, int nE, int nN)
{
    const long long eTot = (long long)nE + nN;
    const long long e = (long long)blockIdx.x * 8 + (threadIdx.x >> 5);
    if (e >= eTot) return;
    const int lane = threadIdx.x & 31;

    int src, dst;
    if (e < nE) { src = ei[e]; dst = ei[(long long)nE + e]; }
    else        { src = (int)(e - nE); dst = src; }

    const int h = lane >> 4;
    const int c = lane & 15;
    const float alpha =
        ebuf[e * 2 + h] / (denom[(long long)dst * 2 + h] + 1e-16f) * 0.5f;
    const float* xl = xlr + (long long)src * IN_CH + h * OUT_CH;
    float* op = out + (long long)dst * OUT_CH;
    atomicAdd(&op[c],      alpha * xl[c]);
    atomicAdd(&op[c + 16], alpha * xl[c + 16]);
}

// ============================================================
// Kernel 5: out = silu(out + bias), in place
// ============================================================
__global__ __launch_bounds__(256) void finalize(
    float* __restrict__ out, const float* __restrict__ bias, int total)
{
    const int t = blockIdx.x * blockDim.x + threadIdx.x;
    if (t >= total) return;
    const float v = out[t] + bias[t & (OUT_CH - 1)];
    out[t] = v / (1.f + __expf(-v));
}

extern "C" void kernel_launch(void* const* d_in, const int* in_sizes, int n_in,
                              void* d_out, int out_size, void* d_ws, size_t ws_size,
                              hipStream_t stream) {
    const float* x    = (const float*)d_in[0];
    const float* Wl   = (const float*)d_in[1];
    const float* bl   = (const float*)d_in[2];
    const float* Wr   = (const float*)d_in[3];
    const float* br   = (const float*)d_in[4];
    const float* att  = (const float*)d_in[5];
    const float* bias = (const float*)d_in[6];
    const int*   ei   = (const int*)d_in[7];

    const int N  = in_sizes[0] / IN_CH;       // 100000
    const int nE = in_sizes[7] / 2;           // 1600000
    const long long eTot = (long long)nE + N; // with self loops

    char* ws = (char*)d_ws;
    size_t off = 0;
    float* xlr = (float*)(ws + off);
    off += (size_t)N * IN_CH * sizeof(float);        off = (off + 255) & ~(size_t)255;
    float* ebuf = (float*)(ws + off);
    off += (size_t)eTot * 2 * sizeof(float);         off = (off + 255) & ~(size_t)255;
    unsigned* mkey = (unsigned*)(ws + off);
    off += (size_t)N * 2 * sizeof(unsigned);         off = (off + 255) & ~(size_t)255;
    float* denom = (float*)(ws + off);

    hipMemsetAsync(mkey,  0, (size_t)N * 2 * sizeof(unsigned), stream);
    hipMemsetAsync(denom, 0, (size_t)N * 2 * sizeof(float), stream);
    hipMemsetAsync(d_out, 0, (size_t)out_size * sizeof(float), stream);

    gemm_xlr<<<N / 16, 256, 0, stream>>>(x, Wl, Wr, bl, br, xlr);

    const int edgeBlocks = (int)((eTot + 7) / 8);   // 8 waves (edges) per block
    edge_logits<<<edgeBlocks, 256, 0, stream>>>(ei, xlr, att, ebuf, mkey, nE, N);

    const long long nT = eTot * 2;
    edge_exp<<<(int)((nT + 255) / 256), 256, 0, stream>>>(ei, ebuf, mkey, denom, nE, N);

    edge_scatter<<<edgeBlocks, 256, 0, stream>>>(ei, xlr, ebuf, denom,
                                                 (float*)d_out, nE, N);

    finalize<<<(N * OUT_CH + 255) / 256, 256, 0, stream>>>((float*)d_out, bias,
                                                           N * OUT_CH);
}